// QOC_22016002359918
// MI455X (gfx1250) — compile-verified
//
#include <hip/hip_runtime.h>

// ---------------------------------------------------------------------------
// QOC: product of 131072 matrix exponentials of 8x8 complex Hamiltonians.
// Strategy: complex 8x8  ->  real 16x16 via R(A)=[[Re,-Im],[Im,Re]] embedding
// (an algebra homomorphism, so expm and chain products commute with it).
// All matmuls run on V_WMMA_F32_16X16X4_F32 (f32 precision == complex64 parts).
// Chunked ordered scan: 1024 waves x 128 steps -> 32 -> 1 -> extract.
// ---------------------------------------------------------------------------

#define N_STEPS   131072
#define CHUNK     128
#define NCHUNK    (N_STEPS / CHUNK)   // 1024
#define COMB      32                  // combine fan-in (1024 -> 32 -> 1)
#define DT_F      1e-4f
#define MAXPOWER_F 20000.0f
#define SSQ       6                   // scaling/squaring: B = A / 2^6
#define INV_SCALE (1.0f / 64.0f)

typedef __attribute__((ext_vector_type(2))) float v2f;
typedef __attribute__((ext_vector_type(8))) float v8f;

// ---- fragment conversion through a 1KB LDS tile -----------------------------
// C/D layout (16x16 f32, 8 VGPRs): lane L, vgpr v -> row = v + 8*(L>=16),
// col = L%16.  A frag (16x4): lane L holds M=L%16, K = k0 + 2*(L>=16) + {0,1}.
// B frag (4x16): lane L holds N=L%16, K = k0 + 2*(L>=16) + {0,1}.
// Single-wave workgroup: __syncthreads() is a wave-level fence (near free),
// and LDS ops are in-order within a wave on CDNA5.

__device__ __forceinline__ void emb_store(const float* X, float* lds, int col, int rh) {
  __syncthreads();
#pragma unroll
  for (int v = 0; v < 8; ++v) lds[(v + 8 * rh) * 16 + col] = X[v];
  __syncthreads();
}

__device__ __forceinline__ void frag_loadA(v2f* fa, const float* lds, int col, int rh) {
#pragma unroll
  for (int j = 0; j < 4; ++j) {
    const int kb = 4 * j + 2 * rh;
    fa[j].x = lds[col * 16 + kb];
    fa[j].y = lds[col * 16 + kb + 1];
  }
}

__device__ __forceinline__ void frag_loadB(v2f* fb, const float* lds, int col, int rh) {
#pragma unroll
  for (int j = 0; j < 4; ++j) {
    const int kb = 4 * j + 2 * rh;
    fb[j].x = lds[kb * 16 + col];
    fb[j].y = lds[(kb + 1) * 16 + col];
  }
}

__device__ __forceinline__ void to_fragA(const float* X, v2f* fa, float* lds, int col, int rh) {
  emb_store(X, lds, col, rh);
  frag_loadA(fa, lds, col, rh);
}
__device__ __forceinline__ void to_fragB(const float* X, v2f* fb, float* lds, int col, int rh) {
  emb_store(X, lds, col, rh);
  frag_loadB(fb, lds, col, rh);
}
__device__ __forceinline__ void to_fragAB(const float* X, v2f* fa, v2f* fb, float* lds, int col, int rh) {
  emb_store(X, lds, col, rh);
  frag_loadA(fa, lds, col, rh);
  frag_loadB(fb, lds, col, rh);
}

// 16x16x16 f32 matmul = 4 accumulating V_WMMA_F32_16X16X4_F32
__device__ __forceinline__ void wmma16(const v2f* fa, const v2f* fb, float* O) {
  v8f acc = {0.f, 0.f, 0.f, 0.f, 0.f, 0.f, 0.f, 0.f};
#pragma unroll
  for (int j = 0; j < 4; ++j)
    acc = __builtin_amdgcn_wmma_f32_16x16x4_f32(false, fa[j], false, fb[j],
                                                (short)0, acc, false, false);
#pragma unroll
  for (int v = 0; v < 8; ++v) O[v] = acc[v];
}

// ---------------------------------------------------------------------------
// Kernel 1: one wave per chunk of 128 steps. Computes ordered product of the
// step propagators in the 16x16 embedding; writes 256 floats per chunk.
// ---------------------------------------------------------------------------
__global__ void __launch_bounds__(32)
qoc_chunk_kernel(const float* __restrict__ a,    // [2, N_STEPS]
                 const float* __restrict__ Hd,   // 8x8 complex64 (128 floats)
                 const float* __restrict__ Hc,   // [2,8,8] complex64 (256 floats)
                 float* __restrict__ part)       // [NCHUNK, 256]
{
  __shared__ float lds[256];
  const int lane = threadIdx.x;
  const int col  = lane & 15;
  const int rh   = lane >> 4;          // row half: rows rh*8 .. rh*8+7

  // Embedding constants for B = (-i*dt/2^SSQ) * (Hd + MAXPOWER*(a0*H1c+a1*H2c)).
  // For s = -i*k, s*(xr + i*xi) has re = k*xi, im = -k*xr.
  // R blocks: TL/BR = re ; TR = -im = k*xr ; BL = im = -k*xr.
  const float kd = DT_F * INV_SCALE;
  const float kc = DT_F * MAXPOWER_F * INV_SCALE;

  float eHd[8], eH1[8], eH2[8], idm[8];
#pragma unroll
  for (int v = 0; v < 8; ++v) {
    const int  r    = v + 8 * rh;
    const int  rr   = r & 7, cc = col & 7;
    const int  idx  = (rr * 8 + cc) * 2;
    const bool rtop = (r < 8), ctop = (col < 8);
    idm[v] = (r == col) ? 1.0f : 0.0f;
    {
      const float xr = Hd[idx], xi = Hd[idx + 1];
      eHd[v] = (rtop == ctop) ? kd * xi : (rtop ? kd * xr : -kd * xr);
    }
    {
      const float xr = Hc[idx], xi = Hc[idx + 1];
      eH1[v] = (rtop == ctop) ? kc * xi : (rtop ? kc * xr : -kc * xr);
    }
    {
      const float xr = Hc[128 + idx], xi = Hc[128 + idx + 1];
      eH2[v] = (rtop == ctop) ? kc * xi : (rtop ? kc * xr : -kc * xr);
    }
  }

  // P = identity (embedding of I8 is I16)
  float P[8];
#pragma unroll
  for (int v = 0; v < 8; ++v) P[v] = idm[v];

  const int i0 = blockIdx.x * CHUNK;
  for (int s = 0; s < CHUNK; ++s) {
    const int   i  = i0 + s;
    const float a0 = a[i];
    const float a1 = a[N_STEPS + i];

    // Scaled matrix B (||B|| <= ~0.2) and Taylor-8 Horner for expm(B).
    float B[8], S[8];
#pragma unroll
    for (int v = 0; v < 8; ++v) {
      B[v] = eHd[v] + a0 * eH1[v] + a1 * eH2[v];
      S[v] = idm[v] + B[v] * 0.125f;          // S = I + B/8
    }
    v2f Bfa[4];
    to_fragA(B, Bfa, lds, col, rh);           // B's A-fragment reused 7x
#pragma unroll
    for (int k = 7; k >= 1; --k) {            // S = I + (B*S)/k
      v2f Sfb[4];
      to_fragB(S, Sfb, lds, col, rh);
      float T[8];
      wmma16(Bfa, Sfb, T);
      const float invk = 1.0f / (float)k;
#pragma unroll
      for (int v = 0; v < 8; ++v) S[v] = idm[v] + T[v] * invk;
    }
    // Undo scaling: square SSQ times.
#pragma unroll
    for (int t = 0; t < SSQ; ++t) {
      v2f fa[4], fb[4];
      to_fragAB(S, fa, fb, lds, col, rh);
      wmma16(fa, fb, S);
    }
    // Ordered chain: P = P @ E
    v2f Pfa[4], Efb[4];
    to_fragA(P, Pfa, lds, col, rh);
    to_fragB(S, Efb, lds, col, rh);
    wmma16(Pfa, Efb, P);
  }

#pragma unroll
  for (int v = 0; v < 8; ++v)
    part[(size_t)blockIdx.x * 256 + (v + 8 * rh) * 16 + col] = P[v];
}

// ---------------------------------------------------------------------------
// Kernel 2: each wave multiplies R consecutive partial products in order.
// ---------------------------------------------------------------------------
__global__ void __launch_bounds__(32)
qoc_combine_kernel(const float* __restrict__ in, float* __restrict__ out, int R)
{
  __shared__ float lds[256];
  const int lane = threadIdx.x;
  const int col  = lane & 15;
  const int rh   = lane >> 4;
  const float* base = in + (size_t)blockIdx.x * R * 256;

  float P[8];
#pragma unroll
  for (int v = 0; v < 8; ++v) P[v] = base[(v + 8 * rh) * 16 + col];

  for (int j = 1; j < R; ++j) {
    float Q[8];
#pragma unroll
    for (int v = 0; v < 8; ++v) Q[v] = base[(size_t)j * 256 + (v + 8 * rh) * 16 + col];
    v2f Pfa[4], Qfb[4];
    to_fragA(P, Pfa, lds, col, rh);
    to_fragB(Q, Qfb, lds, col, rh);
    wmma16(Pfa, Qfb, P);
  }

#pragma unroll
  for (int v = 0; v < 8; ++v)
    out[(size_t)blockIdx.x * 256 + (v + 8 * rh) * 16 + col] = P[v];
}

// ---------------------------------------------------------------------------
// Kernel 3: 16x16 embedding -> 8x8 complex64 (interleaved re,im) output.
// Re(A) = top-left block, Im(A) = bottom-left block.
// ---------------------------------------------------------------------------
__global__ void __launch_bounds__(64)
qoc_extract_kernel(const float* __restrict__ emb, float* __restrict__ out)
{
  const int t  = threadIdx.x;      // 0..63
  const int rr = t >> 3, cc = t & 7;
  out[2 * t + 0] = emb[rr * 16 + cc];         // Re
  out[2 * t + 1] = emb[(rr + 8) * 16 + cc];   // Im
}

// ---------------------------------------------------------------------------
extern "C" void kernel_launch(void* const* d_in, const int* in_sizes, int n_in,
                              void* d_out, int out_size, void* d_ws, size_t ws_size,
                              hipStream_t stream)
{
  (void)in_sizes; (void)n_in; (void)out_size; (void)ws_size;
  const float* a  = (const float*)d_in[0];   // [2, N_STEPS] f32
  const float* Hd = (const float*)d_in[1];   // 8x8 complex64
  const float* Hc = (const float*)d_in[2];   // [2,8,8] complex64

  float* w0 = (float*)d_ws;                          // NCHUNK * 256 floats
  float* w1 = w0 + (size_t)NCHUNK * 256;             // 32 * 256 floats
  float* w2 = w1 + (size_t)COMB * 256;               // 256 floats

  qoc_chunk_kernel<<<NCHUNK, 32, 0, stream>>>(a, Hd, Hc, w0);
  qoc_combine_kernel<<<NCHUNK / COMB, 32, 0, stream>>>(w0, w1, COMB); // 1024 -> 32
  qoc_combine_kernel<<<1, 32, 0, stream>>>(w1, w2, COMB);            // 32 -> 1
  qoc_extract_kernel<<<1, 64, 0, stream>>>(w2, (float*)d_out);
}